// SoftmaxPooling2d_9698036154920
// MI455X (gfx1250) — compile-verified
//
#include <hip/hip_runtime.h>
#include <hip/hip_bf16.h>

// Softmax pooling 2x2, stride 2, no padding, BETA=1.
// x: (32,128,112,112) f32  ->  out: (32,128,56,56) f32
//
// Memory-bound: ~257 MB traffic -> ~11 us floor at 23.3 TB/s.
// Round-2 change: all per-iteration index math is block-uniform (SALU) or
// loop-invariant per-lane constants; VALU does only the softmax arithmetic.

typedef float v4f __attribute__((ext_vector_type(4)));
typedef float v2f __attribute__((ext_vector_type(2)));

namespace {

constexpr int W_IN      = 112;
constexpr int H_OUT     = 56;
constexpr int W_OUT     = 56;
constexpr int PLANES    = 32 * 128;          // N*C = 4096
constexpr int QUADS     = W_OUT / 2;         // 28 float4-pairs per output row
constexpr int IN_PLANE  = 112 * 112;         // 12544 elements
constexpr int OUT_PLANE = H_OUT * W_OUT;     // 3136 elements

// Block-tile = 8 output rows of one plane = 16 input rows = 7168 contiguous B.
constexpr int ROWS_PER_TILE = 8;
constexpr int TILES_PER_PLANE = H_OUT / ROWS_PER_TILE;     // 7
constexpr int NTILES   = PLANES * TILES_PER_PLANE;         // 28672
constexpr int BLOCK    = QUADS * ROWS_PER_TILE;            // 224 = 7 wave32s
constexpr int GRID     = 4096;                             // NTILES/GRID = 7 iters
constexpr int TILE_IN_BYTES = 2 * ROWS_PER_TILE * W_IN * 4;  // 7168
constexpr int PF_LINES = TILE_IN_BYTES / 128;                // 56 cache lines

__device__ __forceinline__ float spool(float a, float b, float c, float d) {
    // softmax over {a,b,c,d} dotted with {a,b,c,d} (max-subtracted, matches
    // jax.nn.softmax numerics).
    float m  = fmaxf(fmaxf(a, b), fmaxf(c, d));
    float ea = __expf(a - m);
    float eb = __expf(b - m);
    float ec = __expf(c - m);
    float ed = __expf(d - m);
    float num = fmaf(a, ea, fmaf(b, eb, fmaf(c, ec, d * ed)));
    float den = ea + eb + ec + ed;          // in (1, 4]
    return __fdividef(num, den);            // v_rcp_f32 + v_mul_f32
}

__global__ __launch_bounds__(BLOCK) void softpool2x2_kernel(
        const float* __restrict__ x, float* __restrict__ out) {
    // Per-lane constants (computed once; loop-invariant):
    const int tid = (int)threadIdx.x;
    const int q   = tid % QUADS;             // float4 index within row
    const int rl  = tid / QUADS;             // row-in-tile [0,8)
    const int in_off  = rl * (2 * W_IN) + 4 * q;   // element offset in tile
    const int out_off = rl * W_OUT + 2 * q;        // element offset in out-tile
    const int pf_off  = tid * 32;                  // 128 B per lane (elements)
    const bool pf_lane = (tid < PF_LINES);

    // Grid-stride over tiles; T is wave-uniform -> SALU index math only.
    for (int T = (int)blockIdx.x; T < NTILES; T += GRID) {
        // ---- prefetch next tile's 56 cache lines into cache (gfx1250) ----
        int Tn = T + GRID;
        if (Tn < NTILES) {
            int pn  = Tn / TILES_PER_PLANE;          // scalar magic-mul
            int rgn = Tn - TILES_PER_PLANE * pn;
            const float* nb = x + (size_t)pn * IN_PLANE
                                + (size_t)rgn * (2 * ROWS_PER_TILE * W_IN);
            if (pf_lane) __builtin_prefetch(nb + pf_off, 0, 1);
        }

        // ---- current tile ----
        int p  = T / TILES_PER_PLANE;                // scalar
        int rg = T - TILES_PER_PLANE * p;            // scalar

        const float* tile = x + (size_t)p * IN_PLANE
                              + (size_t)rg * (2 * ROWS_PER_TILE * W_IN);

        v4f a = __builtin_nontemporal_load((const v4f*)(tile + in_off));        // row 2*oy
        v4f b = __builtin_nontemporal_load((const v4f*)(tile + in_off + W_IN)); // row 2*oy+1

        v2f o;
        o.x = spool(a.x, a.y, b.x, b.y);       // window at ox = 2q
        o.y = spool(a.z, a.w, b.z, b.w);       // window at ox = 2q+1

        float* dst = out + (size_t)p * OUT_PLANE
                         + (size_t)rg * (ROWS_PER_TILE * W_OUT) + out_off;
        __builtin_nontemporal_store(o, (v2f*)dst);   // 8-B aligned b64 store
    }
}

} // anonymous namespace

extern "C" void kernel_launch(void* const* d_in, const int* in_sizes, int n_in,
                              void* d_out, int out_size, void* d_ws, size_t ws_size,
                              hipStream_t stream) {
    (void)in_sizes; (void)n_in; (void)d_ws; (void)ws_size; (void)out_size;
    const float* x = (const float*)d_in[0];
    float* out = (float*)d_out;
    // 4096 blocks x 224 threads (7 wave32s) -> 28672 waves, exactly 7
    // grid-stride iterations per block, prefetch one stride (~29 MB) ahead.
    softpool2x2_kernel<<<GRID, BLOCK, 0, stream>>>(x, out);
}